// ConfigurableGATEncoder_16114717294846
// MI455X (gfx1250) — compile-verified
//
#include <hip/hip_runtime.h>
#include <cmath>

typedef __attribute__((ext_vector_type(2))) float v2f;
typedef __attribute__((ext_vector_type(8))) float v8f;

static constexpr float kNegSlope = 0.2f;

// ---------------------------------------------------------------- utilities
__global__ void fill_f32(float* __restrict__ p, float v, long n) {
  long i = (long)blockIdx.x * blockDim.x + threadIdx.x;
  if (i < n) p[i] = v;
}

// ------------------------------------------------- h0 = [x[:, :32], emb[id]]
__global__ void build_h0(const float* __restrict__ x, const float* __restrict__ emb,
                         float* __restrict__ h0, int N) {
  int n = blockIdx.x * blockDim.x + threadIdx.x;
  if (n >= N) return;
  const float* xr = x + (long)n * 33;
  float* hr = h0 + (long)n * 48;
#pragma unroll
  for (int i = 0; i < 32; ++i) hr[i] = xr[i];
  int cid = (int)xr[32];
  const float* er = emb + (long)cid * 16;
#pragma unroll
  for (int i = 0; i < 16; ++i) hr[32 + i] = er[i];
}

// ------------------------------------------------------------- f32 WMMA GEMM
// C(MxNCOLS) = A(MxK) * B(KxNCOLS), all row-major, M a multiple of 16.
// One wave per 16x16 output tile; blockDim.x = 32 * (NCOLS/16).
// A-matrix 16x4 f32 layout: lane = M (0..15 repeated), half-wave picks K pair.
// C/D layout: VGPR r -> row r + 8*(lane>=16), col = lane & 15.
template <int K, int NCOLS>
__global__ void wmma_gemm(const float* __restrict__ A, const float* __restrict__ B,
                          float* __restrict__ C) {
  const int lane  = threadIdx.x & 31;
  const int nTile = threadIdx.x >> 5;            // wave id -> n tile
  const int mTile = blockIdx.x;
  const int mRow  = (mTile << 4) + (lane & 15);  // A row for this lane
  const int nCol  = (nTile << 4) + (lane & 15);  // B/C column for this lane
  const int kHalf = (lane >> 4) << 1;            // 0 or 2

  v8f acc = {};
#pragma unroll
  for (int k0 = 0; k0 < K; k0 += 4) {
    v2f a, b;
    a.x = A[(long)mRow * K + (k0 + kHalf)];
    a.y = A[(long)mRow * K + (k0 + kHalf + 1)];
    b.x = B[(long)(k0 + kHalf) * NCOLS + nCol];
    b.y = B[(long)(k0 + kHalf + 1) * NCOLS + nCol];
    acc = __builtin_amdgcn_wmma_f32_16x16x4_f32(
        /*neg_a=*/false, a, /*neg_b=*/false, b,
        /*c_mod=*/(short)0, acc, /*reuse_a=*/false, /*reuse_b=*/false);
  }
  const int rowBase = (mTile << 4) + ((lane >> 4) << 3);
#pragma unroll
  for (int r = 0; r < 8; ++r)
    C[(long)(rowBase + r) * NCOLS + nCol] = acc[r];
}

// ------------------------------------------ per-(node,head) attention logits
template <int H>
__global__ void node_alpha(const float* __restrict__ h, const float* __restrict__ a_src,
                           const float* __restrict__ a_dst, float* __restrict__ as,
                           float* __restrict__ ad, int N) {
  int i = blockIdx.x * blockDim.x + threadIdx.x;  // i = n*H + hh
  if (i >= N * H) return;
  int hh = i % H;
  const float* hr  = h + (long)i * 64;
  const float* asr = a_src + hh * 64;
  const float* adr = a_dst + hh * 64;
  float s = 0.f, d = 0.f;
#pragma unroll
  for (int c = 0; c < 64; ++c) { s += hr[c] * asr[c]; d += hr[c] * adr[c]; }
  as[i] = s;
  ad[i] = d;
}

// --------------------------------------------------------------- edge passes
template <int H>
__device__ __forceinline__ void edge_sd(const int* ei, long e, int E, int& src, int& dst) {
  if (e < E) { src = ei[e]; dst = ei[(long)E + e]; }
  else       { src = dst = (int)(e - E); }
}

template <int H>
__global__ void edge_max(const int* __restrict__ ei, int E, int N,
                         const float* __restrict__ as, const float* __restrict__ ad,
                         float* __restrict__ m) {
  long t = (long)blockIdx.x * blockDim.x + threadIdx.x;
  long nE = (long)E + N;
  if (t >= nE * H) return;
  long e = t / H; int hh = (int)(t % H);
  int src, dst; edge_sd<H>(ei, e, E, src, dst);
  float v = as[(long)src * H + hh] + ad[(long)dst * H + hh];
  v = v > 0.f ? v : kNegSlope * v;
  atomicMax(&m[(long)dst * H + hh], v);
}

template <int H>
__global__ void edge_expsum(const int* __restrict__ ei, int E, int N,
                            const float* __restrict__ as, const float* __restrict__ ad,
                            const float* __restrict__ m, float* __restrict__ den) {
  long t = (long)blockIdx.x * blockDim.x + threadIdx.x;
  long nE = (long)E + N;
  if (t >= nE * H) return;
  long e = t / H; int hh = (int)(t % H);
  int src, dst; edge_sd<H>(ei, e, E, src, dst);
  float v = as[(long)src * H + hh] + ad[(long)dst * H + hh];
  v = v > 0.f ? v : kNegSlope * v;
  atomicAdd(&den[(long)dst * H + hh], expf(v - m[(long)dst * H + hh]));
}

// one wave per edge: gather h[src], scatter-add alpha*h[src] into acc[dst]
template <int H>
__global__ void edge_message(const int* __restrict__ ei, int E, int N,
                             const float* __restrict__ as, const float* __restrict__ ad,
                             const float* __restrict__ m, const float* __restrict__ den,
                             const float* __restrict__ h, float* __restrict__ acc) {
  int lane = threadIdx.x & 31;
  long e = (long)blockIdx.x * (blockDim.x >> 5) + (threadIdx.x >> 5);
  long nE = (long)E + N;
  if (e >= nE) return;
  int src, dst; edge_sd<H>(ei, e, E, src, dst);
#pragma unroll
  for (int hh = 0; hh < H; ++hh) {
    float v = as[(long)src * H + hh] + ad[(long)dst * H + hh];
    v = v > 0.f ? v : kNegSlope * v;
    float a = expf(v - m[(long)dst * H + hh]) / den[(long)dst * H + hh];
    const float* hs = h + ((long)src * H + hh) * 64;
    float* ac = acc + ((long)dst * H + hh) * 64;
    atomicAdd(&ac[lane],      hs[lane]      * a);
    atomicAdd(&ac[lane + 32], hs[lane + 32] * a);
  }
}

// -------------------------- mean(heads) + b0, LayerNorm(64), ELU  (wave/node)
__global__ void finalize0(const float* __restrict__ acc, const float* __restrict__ b0,
                          const float* __restrict__ g, const float* __restrict__ bln,
                          float* __restrict__ h1, int N) {
  int lane = threadIdx.x & 31;
  int n = blockIdx.x * (blockDim.x >> 5) + (threadIdx.x >> 5);
  if (n >= N) return;
  const float* a0 = acc + (long)n * 128;  // (2 heads, 64)
  float v0 = 0.5f * (a0[lane]      + a0[64 + lane]) + b0[lane];
  float v1 = 0.5f * (a0[lane + 32] + a0[96 + lane]) + b0[lane + 32];
  float s = v0 + v1;
#pragma unroll
  for (int off = 16; off; off >>= 1) s += __shfl_xor(s, off, 32);
  float mu = s * (1.f / 64.f);
  float d0 = v0 - mu, d1 = v1 - mu;
  float q = d0 * d0 + d1 * d1;
#pragma unroll
  for (int off = 16; off; off >>= 1) q += __shfl_xor(q, off, 32);
  float rstd = rsqrtf(q * (1.f / 64.f) + 1e-5f);
  float o0 = d0 * rstd * g[lane]      + bln[lane];
  float o1 = d1 * rstd * g[lane + 32] + bln[lane + 32];
  o0 = o0 > 0.f ? o0 : expm1f(o0);  // ELU
  o1 = o1 > 0.f ? o1 : expm1f(o1);
  h1[(long)n * 64 + lane]      = o0;
  h1[(long)n * 64 + lane + 32] = o1;
}

__global__ void add_bias(float* __restrict__ out, const float* __restrict__ b, int N) {
  int i = blockIdx.x * blockDim.x + threadIdx.x;
  if (i < N * 64) out[i] += b[i & 63];
}

// =========================================================================
extern "C" void kernel_launch(void* const* d_in, const int* in_sizes, int n_in,
                              void* d_out, int out_size, void* d_ws, size_t ws_size,
                              hipStream_t stream) {
  const float* x      = (const float*)d_in[0];
  const int*   ei     = (const int*)  d_in[1];
  const float* emb    = (const float*)d_in[2];
  const float* W0     = (const float*)d_in[3];
  const float* a_src0 = (const float*)d_in[4];
  const float* a_dst0 = (const float*)d_in[5];
  const float* b0     = (const float*)d_in[6];
  const float* ln_g   = (const float*)d_in[7];
  const float* ln_b   = (const float*)d_in[8];
  const float* W1     = (const float*)d_in[9];
  const float* a_src1 = (const float*)d_in[10];
  const float* a_dst1 = (const float*)d_in[11];
  const float* b1     = (const float*)d_in[12];
  float* out = (float*)d_out;

  const int  N  = in_sizes[0] / 33;  // 50000 (divisible by 16)
  const int  E  = in_sizes[1] / 2;   // 800000
  const long nE = (long)E + N;       // + self loops

  // workspace bump allocator
  char* ws = (char*)d_ws;
  size_t off = 0;
  auto walloc = [&](size_t bytes) -> float* {
    float* p = (float*)(ws + off);
    off += (bytes + 255) & ~(size_t)255;
    return p;
  };
  float* h0   = walloc((size_t)N * 48  * 4);
  float* hL0  = walloc((size_t)N * 128 * 4);   // layer0 features (N,2,64)
  float* as0  = walloc((size_t)N * 2 * 4);
  float* ad0  = walloc((size_t)N * 2 * 4);
  float* m0   = walloc((size_t)N * 2 * 4);
  float* den0 = walloc((size_t)N * 2 * 4);
  float* acc0 = walloc((size_t)N * 128 * 4);
  float* h1   = walloc((size_t)N * 64 * 4);
  float* h2   = walloc((size_t)N * 64 * 4);    // layer1 features (N,1,64)
  float* as1  = walloc((size_t)N * 4);
  float* ad1  = walloc((size_t)N * 4);
  float* m1   = walloc((size_t)N * 4);
  float* den1 = walloc((size_t)N * 4);
  (void)ws_size; (void)n_in; (void)out_size;

  const int T = 256;
  // ---------------- layer 0 ----------------
  build_h0<<<(N + T - 1) / T, T, 0, stream>>>(x, emb, h0, N);
  wmma_gemm<48, 128><<<N / 16, 256, 0, stream>>>(h0, W0, hL0);
  node_alpha<2><<<(N * 2 + T - 1) / T, T, 0, stream>>>(hL0, a_src0, a_dst0, as0, ad0, N);
  fill_f32<<<(int)(((long)N * 2 + T - 1) / T), T, 0, stream>>>(m0, -INFINITY, (long)N * 2);
  fill_f32<<<(int)(((long)N * 2 + T - 1) / T), T, 0, stream>>>(den0, 0.f, (long)N * 2);
  fill_f32<<<(int)(((long)N * 128 + T - 1) / T), T, 0, stream>>>(acc0, 0.f, (long)N * 128);
  {
    int gb = (int)((nE * 2 + T - 1) / T);
    edge_max<2><<<gb, T, 0, stream>>>(ei, E, N, as0, ad0, m0);
    edge_expsum<2><<<gb, T, 0, stream>>>(ei, E, N, as0, ad0, m0, den0);
    edge_message<2><<<(int)((nE + 7) / 8), T, 0, stream>>>(ei, E, N, as0, ad0, m0, den0,
                                                           hL0, acc0);
  }
  finalize0<<<(N + 7) / 8, T, 0, stream>>>(acc0, b0, ln_g, ln_b, h1, N);

  // ---------------- layer 1 ----------------
  wmma_gemm<64, 64><<<N / 16, 128, 0, stream>>>(h1, W1, h2);
  node_alpha<1><<<(N + T - 1) / T, T, 0, stream>>>(h2, a_src1, a_dst1, as1, ad1, N);
  fill_f32<<<(int)(((long)N + T - 1) / T), T, 0, stream>>>(m1, -INFINITY, (long)N);
  fill_f32<<<(int)(((long)N + T - 1) / T), T, 0, stream>>>(den1, 0.f, (long)N);
  fill_f32<<<(int)(((long)N * 64 + T - 1) / T), T, 0, stream>>>(out, 0.f, (long)N * 64);
  {
    int gb = (int)((nE + T - 1) / T);
    edge_max<1><<<gb, T, 0, stream>>>(ei, E, N, as1, ad1, m1);
    edge_expsum<1><<<gb, T, 0, stream>>>(ei, E, N, as1, ad1, m1, den1);
    edge_message<1><<<(int)((nE + 7) / 8), T, 0, stream>>>(ei, E, N, as1, ad1, m1, den1,
                                                           h2, out);
  }
  add_bias<<<(N * 64 + T - 1) / T, T, 0, stream>>>(out, b1, N);
}